// DenseEdgeGATLayer_18528488915214
// MI455X (gfx1250) — compile-verified
//
#include <hip/hip_runtime.h>
#include <hip/hip_bf16.h>
#include <stdint.h>

typedef float v2f __attribute__((ext_vector_type(2)));
typedef float v8f __attribute__((ext_vector_type(8)));

#define N_NODES 1024
#define DIM 128
#define HEADS 8
#define DK 16
#define EDIM 16
#define TN 16                       // target nodes per WG
#define MCHUNK 256                  // source nodes per WG
#define NCHUNKS (N_NODES / MCHUNK)  // 4
#define MTILES (MCHUNK / 16)        // 16

static __device__ __forceinline__ v8f wmma4(v2f a, v2f b, v8f c) {
    return __builtin_amdgcn_wmma_f32_16x16x4_f32(false, a, false, b, (short)0, c,
                                                 false, false);
}

// ---------------------------------------------------------------------------
// Kernel 1: row LayerNorm. 1024 blocks x 128 threads, one row per block.
// ---------------------------------------------------------------------------
__global__ void ln_kernel(const float* __restrict__ x,
                          const float* __restrict__ gamma,
                          const float* __restrict__ beta,
                          float* __restrict__ h) {
    __shared__ float red[DIM];
    const int row = blockIdx.x;
    const int t = threadIdx.x;
    float v = x[row * DIM + t];
    red[t] = v;
    __syncthreads();
    for (int s = 64; s > 0; s >>= 1) {
        if (t < s) red[t] += red[t + s];
        __syncthreads();
    }
    float mu = red[0] * (1.0f / DIM);
    __syncthreads();
    float d = v - mu;
    red[t] = d * d;
    __syncthreads();
    for (int s = 64; s > 0; s >>= 1) {
        if (t < s) red[t] += red[t + s];
        __syncthreads();
    }
    float var = red[0] * (1.0f / DIM);
    float r = rsqrtf(var + 1e-5f);
    h[row * DIM + t] = d * r * gamma[t] + beta[t];
}

// ---------------------------------------------------------------------------
// fp32 WMMA GEMM  C(1024x128) = A(1024x128) @ W(128x128) + bias (+ residual).
// A-frag (16x4): lane<16 -> M=lane, K=k0+{0,1}; lane>=16 -> M=lane-16, K=k0+{2,3}
// B-frag (4x16): lane<16 -> N=lane, K=k0+{0,1}; lane>=16 -> N=lane-16, K=k0+{2,3}
// C/D: VGPR r, lane<16 -> (M=r, N=lane); lane>=16 -> (M=8+r, N=lane-16)
// ---------------------------------------------------------------------------
__global__ __launch_bounds__(256) void wmma_gemm128(const float* __restrict__ A,
                                                    const float* __restrict__ W,
                                                    const float* __restrict__ bias,
                                                    const float* __restrict__ resid,
                                                    float* __restrict__ C) {
    const int wave = threadIdx.x >> 5;
    const int lane = threadIdx.x & 31;
    const int row0 = blockIdx.x * 16;
    const int col0 = wave * 16;
    const int m = lane & 15;
    const int kk = (lane >> 4) << 1;

    v8f acc = {};
    #pragma unroll
    for (int k0 = 0; k0 < DIM; k0 += 4) {
        v2f a, b;
        a.x = A[(row0 + m) * DIM + k0 + kk];
        a.y = A[(row0 + m) * DIM + k0 + kk + 1];
        b.x = W[(k0 + kk) * DIM + col0 + m];
        b.y = W[(k0 + kk + 1) * DIM + col0 + m];
        acc = wmma4(a, b, acc);
    }
    const int n = col0 + (lane & 15);
    const int mbase = row0 + ((lane >> 4) << 3);
    const float bb = bias[n];
    #pragma unroll
    for (int r = 0; r < 8; ++r) {
        const int rr = mbase + r;
        float val = acc[r] + bb;
        if (resid) val += resid[rr * DIM + n];
        C[rr * DIM + n] = val;
    }
}

// ---------------------------------------------------------------------------
// Attention partials: blockIdx.x = 16-node tile, blockIdx.y = 256-source chunk.
// 8 waves; wave w owns head w for logits / softmax stats / alpha@V, and target
// nodes {2w, 2w+1} for the alpha^T @ edge_feat aggregation.
// ---------------------------------------------------------------------------
__device__ __forceinline__ void stage_tile(const float* __restrict__ ef,
                                           const uint8_t* __restrict__ mask,
                                           float* s_ef, float* s_maskf,
                                           int t, int n0, int m0) {
    const int nn = t >> 4, mm = t & 15;
    const float4* src =
        (const float4*)(ef + ((size_t)(n0 + nn) * N_NODES + (m0 + mm)) * EDIM);
    float4* dst = (float4*)(s_ef + (nn * 16 + mm) * EDIM);
    #pragma unroll
    for (int i = 0; i < 4; ++i) dst[i] = src[i];
    if (t < 64) {
        const int nb = t >> 2, j0 = (t & 3) * 4;
        #pragma unroll
        for (int j = 0; j < 4; ++j)
            s_maskf[nb * 16 + j0 + j] =
                mask[(size_t)(n0 + nb) * N_NODES + m0 + j0 + j] ? 1.0f : 0.0f;
    }
}

__device__ __forceinline__ void bias_tile(const float* s_ef, const float* s_waep,
                                          float* s_bias, int w, int lh, int hi,
                                          int kk, float baev) {
    #pragma unroll
    for (int half = 0; half < 2; ++half) {
        const int nn = w + half * 8;
        v8f c;
        #pragma unroll
        for (int r = 0; r < 8; ++r) c[r] = baev;
        #pragma unroll
        for (int i = 0; i < 4; ++i) {
            const int e0 = 4 * i + kk;
            v2f a, b;
            a.x = s_ef[(nn * 16 + lh) * EDIM + e0];
            a.y = s_ef[(nn * 16 + lh) * EDIM + e0 + 1];
            b.x = s_waep[e0 * 16 + lh];
            b.y = s_waep[(e0 + 1) * 16 + lh];
            c = wmma4(a, b, c);
        }
        #pragma unroll
        for (int r = 0; r < 8; ++r) {
            const int mm = r + hi * 8;
            if (lh < HEADS) s_bias[(nn * 16 + mm) * HEADS + lh] = c[r];
        }
    }
}

__device__ __forceinline__ v8f logits_tile(const float* __restrict__ kmat,
                                           const float* s_bias, const v2f* qa,
                                           int w, int lh, int hi, int kk, int m0) {
    v8f s;
    #pragma unroll
    for (int r = 0; r < 8; ++r)
        s[r] = s_bias[((r + hi * 8) * 16 + lh) * HEADS + w];
    #pragma unroll
    for (int i = 0; i < 4; ++i) {
        const int d = 4 * i + kk;
        v2f b;
        b.x = kmat[(m0 + lh) * DIM + w * DK + d];
        b.y = kmat[(m0 + lh) * DIM + w * DK + d + 1];
        s = wmma4(qa[i], b, s);
    }
    return s;
}

__global__ __launch_bounds__(256) void att_part_kernel(
    const float* __restrict__ q, const float* __restrict__ k,
    const float* __restrict__ v, const float* __restrict__ ef,
    const uint8_t* __restrict__ mask, const float* __restrict__ Wae,
    const float* __restrict__ bae, float* __restrict__ part_v,
    float* __restrict__ part_w, float* __restrict__ part_mx,
    float* __restrict__ part_sm) {
    __shared__ float s_ef[TN * 16 * EDIM];     // [n][mm][e]
    __shared__ float s_bias[TN * 16 * HEADS];  // [n][mm][h]
    __shared__ float s_alpha[16 * TN * 16];    // [h padded to 16][n][mm]
    __shared__ float s_maskf[TN * 16];
    __shared__ float s_waep[EDIM * 16];        // Wae zero-padded to 16 cols

    const int t = threadIdx.x;
    const int w = t >> 5;        // wave == head
    const int lane = t & 31;
    const int lh = lane & 15;
    const int hi = lane >> 4;
    const int kk = hi << 1;
    const int n0 = blockIdx.x * TN;
    const int ch = blockIdx.y;
    const int mbase = ch * MCHUNK;
    const int PB = blockIdx.x * NCHUNKS + ch;

    s_waep[t] = ((t & 15) < HEADS) ? Wae[(t >> 4) * HEADS + (t & 15)] : 0.0f;
    #pragma unroll
    for (int i = 0; i < 8; ++i) s_alpha[8 * TN * 16 + i * 256 + t] = 0.0f;

    // preload q fragments for this head, pre-scaled by 1/sqrt(dk)
    v2f qa[4];
    #pragma unroll
    for (int i = 0; i < 4; ++i) {
        const int d = 4 * i + kk;
        qa[i].x = 0.25f * q[(n0 + lh) * DIM + w * DK + d];
        qa[i].y = 0.25f * q[(n0 + lh) * DIM + w * DK + d + 1];
    }
    const float baev = (lh < HEADS) ? bae[lh] : 0.0f;

    float rmax[8];
    #pragma unroll
    for (int r = 0; r < 8; ++r) rmax[r] = -3.0e38f;

    // ---- pass 1: per-(n,h) running max over this chunk ----
    for (int mt = 0; mt < MTILES; ++mt) {
        const int m0 = mbase + mt * 16;
        __syncthreads();
        stage_tile(ef, mask, s_ef, s_maskf, t, n0, m0);
        __syncthreads();
        bias_tile(s_ef, s_waep, s_bias, w, lh, hi, kk, baev);
        __syncthreads();
        v8f s = logits_tile(k, s_bias, qa, w, lh, hi, kk, m0);
        #pragma unroll
        for (int r = 0; r < 8; ++r) {
            const float mb = s_maskf[(r + hi * 8) * 16 + lh];
            const float sv = (mb != 0.0f) ? s[r] : -10000.0f;
            rmax[r] = fmaxf(rmax[r], sv);
        }
    }
    #pragma unroll
    for (int r = 0; r < 8; ++r) {
        #pragma unroll
        for (int sh = 1; sh < 16; sh <<= 1)
            rmax[r] = fmaxf(rmax[r], __shfl_xor(rmax[r], sh, 32));
    }

    // ---- pass 2: exp/sum + WMMA aggregations (unnormalized) ----
    float rsum[8];
    #pragma unroll
    for (int r = 0; r < 8; ++r) rsum[r] = 0.0f;
    v8f vacc = {};
    v8f wacc0 = {}, wacc1 = {};
    const int na0 = 2 * w, na1 = 2 * w + 1;

    for (int mt = 0; mt < MTILES; ++mt) {
        const int m0 = mbase + mt * 16;
        __syncthreads();
        stage_tile(ef, mask, s_ef, s_maskf, t, n0, m0);
        __syncthreads();
        bias_tile(s_ef, s_waep, s_bias, w, lh, hi, kk, baev);
        __syncthreads();
        v8f s = logits_tile(k, s_bias, qa, w, lh, hi, kk, m0);
        #pragma unroll
        for (int r = 0; r < 8; ++r) {
            const int nn = r + hi * 8;
            const float mb = s_maskf[nn * 16 + lh];
            float e_ = 0.0f;
            if (mb != 0.0f) e_ = __expf(s[r] - rmax[r]);
            rsum[r] += e_;
            s_alpha[(w * TN + nn) * 16 + lh] = e_;
        }
        __syncthreads();
        // alpha(16n x 16m) @ V_h(16m x 16d), accumulate in registers
        #pragma unroll
        for (int i = 0; i < 4; ++i) {
            const int mm = 4 * i + kk;
            v2f a, b;
            a.x = s_alpha[(w * TN + lh) * 16 + mm];
            a.y = s_alpha[(w * TN + lh) * 16 + mm + 1];
            b.x = v[(m0 + mm) * DIM + w * DK + lh];
            b.y = v[(m0 + mm + 1) * DIM + w * DK + lh];
            vacc = wmma4(a, b, vacc);
        }
        // alpha^T(16h x 16m) @ ef_n(16m x 16e) for n = na0, na1 (rows h>=8 are 0)
        #pragma unroll
        for (int i = 0; i < 4; ++i) {
            const int mm = 4 * i + kk;
            v2f a, b;
            a.x = s_alpha[(lh * TN + na0) * 16 + mm];
            a.y = s_alpha[(lh * TN + na0) * 16 + mm + 1];
            b.x = s_ef[(na0 * 16 + mm) * EDIM + lh];
            b.y = s_ef[(na0 * 16 + mm + 1) * EDIM + lh];
            wacc0 = wmma4(a, b, wacc0);
            a.x = s_alpha[(lh * TN + na1) * 16 + mm];
            a.y = s_alpha[(lh * TN + na1) * 16 + mm + 1];
            b.x = s_ef[(na1 * 16 + mm) * EDIM + lh];
            b.y = s_ef[(na1 * 16 + mm + 1) * EDIM + lh];
            wacc1 = wmma4(a, b, wacc1);
        }
    }

    #pragma unroll
    for (int r = 0; r < 8; ++r) {
        #pragma unroll
        for (int sh = 1; sh < 16; sh <<= 1) rsum[r] += __shfl_xor(rsum[r], sh, 32);
    }

    if (lh == 0) {
        #pragma unroll
        for (int r = 0; r < 8; ++r) {
            const int nn = r + hi * 8;
            part_mx[(size_t)PB * (TN * HEADS) + nn * HEADS + w] = rmax[r];
            part_sm[(size_t)PB * (TN * HEADS) + nn * HEADS + w] = rsum[r];
        }
    }
    #pragma unroll
    for (int r = 0; r < 8; ++r)
        part_v[(size_t)PB * (TN * DIM) + (r + hi * 8) * DIM + w * DK + lh] = vacc[r];
    if (hi == 0) {
        #pragma unroll
        for (int r = 0; r < 8; ++r) {
            part_w[(size_t)PB * (TN * DIM) + na0 * DIM + r * DK + lh] = wacc0[r];
            part_w[(size_t)PB * (TN * DIM) + na1 * DIM + r * DK + lh] = wacc1[r];
        }
    }
}

// ---------------------------------------------------------------------------
// Combine chunk partials (log-sum-exp merge), fold in Wve, write out_pre.
// ---------------------------------------------------------------------------
__global__ __launch_bounds__(256) void att_combine_kernel(
    const float* __restrict__ part_v, const float* __restrict__ part_w,
    const float* __restrict__ part_mx, const float* __restrict__ part_sm,
    const float* __restrict__ Wve, const float* __restrict__ bve,
    float* __restrict__ op) {
    __shared__ float s_f[NCHUNKS * TN * HEADS];
    __shared__ float s_is[TN * HEADS];
    __shared__ float s_w[TN * DIM];
    const int t = threadIdx.x;
    const int nt = blockIdx.x;
    if (t < TN * HEADS) {
        float M = -3.0e38f;
        for (int ch = 0; ch < NCHUNKS; ++ch)
            M = fmaxf(M, part_mx[(size_t)(nt * NCHUNKS + ch) * (TN * HEADS) + t]);
        float S = 0.0f;
        for (int ch = 0; ch < NCHUNKS; ++ch) {
            const float f =
                __expf(part_mx[(size_t)(nt * NCHUNKS + ch) * (TN * HEADS) + t] - M);
            s_f[ch * (TN * HEADS) + t] = f;
            S += f * part_sm[(size_t)(nt * NCHUNKS + ch) * (TN * HEADS) + t];
        }
        s_is[t] = 1.0f / S;
    }
    __syncthreads();
    #pragma unroll
    for (int j = 0; j < 8; ++j) {
        const int idx = t + j * 256;
        const int nn = idx >> 7, c = idx & 127, hh = c >> 4;
        float acc = 0.0f;
        for (int ch = 0; ch < NCHUNKS; ++ch)
            acc += part_w[(size_t)(nt * NCHUNKS + ch) * (TN * DIM) + idx] *
                   s_f[ch * (TN * HEADS) + nn * HEADS + hh];
        s_w[idx] = acc * s_is[nn * HEADS + hh];
    }
    __syncthreads();
    #pragma unroll
    for (int j = 0; j < 8; ++j) {
        const int idx = t + j * 256;
        const int nn = idx >> 7, c = idx & 127, hh = c >> 4;
        float va = 0.0f;
        for (int ch = 0; ch < NCHUNKS; ++ch)
            va += part_v[(size_t)(nt * NCHUNKS + ch) * (TN * DIM) + idx] *
                  s_f[ch * (TN * HEADS) + nn * HEADS + hh];
        va *= s_is[nn * HEADS + hh];
        float oe = 0.0f;
        #pragma unroll
        for (int e = 0; e < EDIM; ++e)
            oe += s_w[nn * DIM + hh * DK + e] * Wve[e * DIM + c];
        op[(size_t)(nt * TN + nn) * DIM + c] = va + oe + bve[c];
    }
}

// ---------------------------------------------------------------------------
extern "C" void kernel_launch(void* const* d_in, const int* in_sizes, int n_in,
                              void* d_out, int out_size, void* d_ws, size_t ws_size,
                              hipStream_t stream) {
    const float*   x    = (const float*)d_in[0];
    const float*   ef   = (const float*)d_in[1];
    const uint8_t* mask = (const uint8_t*)d_in[2];
    const float*   Wq   = (const float*)d_in[3];
    const float*   bq   = (const float*)d_in[4];
    const float*   Wk   = (const float*)d_in[5];
    const float*   bk   = (const float*)d_in[6];
    const float*   Wv   = (const float*)d_in[7];
    const float*   bv   = (const float*)d_in[8];
    const float*   Wae  = (const float*)d_in[9];
    const float*   bae  = (const float*)d_in[10];
    const float*   Wve  = (const float*)d_in[11];
    const float*   bve  = (const float*)d_in[12];
    const float*   Wo   = (const float*)d_in[13];
    const float*   bo   = (const float*)d_in[14];
    const float*   gamma= (const float*)d_in[15];
    const float*   beta = (const float*)d_in[16];
    float* out = (float*)d_out;

    // workspace layout (floats)
    float* ws      = (float*)d_ws;
    float* h       = ws;                          // 1024*128
    float* q       = h + N_NODES * DIM;
    float* k       = q + N_NODES * DIM;
    float* v       = k + N_NODES * DIM;
    float* op      = v + N_NODES * DIM;
    float* part_v  = op + N_NODES * DIM;          // 256 * 2048
    float* part_w  = part_v + (64 * NCHUNKS) * (TN * DIM);
    float* part_mx = part_w + (64 * NCHUNKS) * (TN * DIM);   // 256 * 128
    float* part_sm = part_mx + (64 * NCHUNKS) * (TN * HEADS);

    ln_kernel<<<N_NODES, DIM, 0, stream>>>(x, gamma, beta, h);

    wmma_gemm128<<<N_NODES / 16, 256, 0, stream>>>(h, Wq, bq, nullptr, q);
    wmma_gemm128<<<N_NODES / 16, 256, 0, stream>>>(h, Wk, bk, nullptr, k);
    wmma_gemm128<<<N_NODES / 16, 256, 0, stream>>>(h, Wv, bv, nullptr, v);

    att_part_kernel<<<dim3(N_NODES / TN, NCHUNKS), 256, 0, stream>>>(
        q, k, v, ef, mask, Wae, bae, part_v, part_w, part_mx, part_sm);

    att_combine_kernel<<<N_NODES / TN, 256, 0, stream>>>(
        part_v, part_w, part_mx, part_sm, Wve, bve, op);

    wmma_gemm128<<<N_NODES / 16, 256, 0, stream>>>(op, Wo, bo, x, out);
}